// LRTLinear_40003325395299
// MI455X (gfx1250) — compile-verified
//
#include <hip/hip_runtime.h>

// Rank-64 reformulation of the TT layer:
//   out = (x_pad @ U) @ V + bias,  U = c0*c1*c2*c3 (4096x64), V = c4*c5*c6*c7 (64x4096)
// fp32 end-to-end using V_WMMA_F32_16X16X4_F32 (memory-bound op; ~262MB traffic).
// K of the x@U GEMM is split 4 ways (2048 waves) for latency hiding; partials are
// reduced in a fixed order (deterministic) directly into the packed A-operand layout.

typedef __attribute__((ext_vector_type(2))) float v2f;
typedef __attribute__((ext_vector_type(8))) float v8f;

__device__ __forceinline__ v8f wmma4(v2f a, v2f b, v8f c) {
  // 8-arg f32 form: (neg_a, A, neg_b, B, c_mod, C, reuse_a, reuse_b)
  return __builtin_amdgcn_wmma_f32_16x16x4_f32(false, a, false, b, (short)0, c,
                                               false, false);
}

// ---------------- workspace layout (float offsets) ----------------
#define WS_T1   0          //  4096    T1[d0][d1][r2]
#define WS_T2   4096       // 32768    T2[d0d1][d2][r3]
#define WS_U    36864      // 262144   U[i][r4]      (4096x64)
#define WS_Q2   299008     //  4096    Q2[s5][d6d7]
#define WS_Q3   303104     // 32768    Q3[s4][d5d6d7]
#define WS_V    335872     // 262144   V[r4][j]      (64x4096)
#define WS_UPK  598016     // 256000   packed B-operand for x@U (1000 K-steps x 4 ntiles)
#define WS_VPK  854016     // 262144   packed B-operand for Z@V (256 ntiles x 16 K-steps)
#define WS_ZP   1116160    // 4*524288 partial Z (4 x 8192x64, plain row-major)
#define WS_ZPK  3213312    // 524288   Z packed as WMMA A-operand (512 mtiles x 16 t x 32 lanes x 2)

// ---------------- tiny prep kernels: compose cores into U and V ----------------
__global__ void k_t1(const float* __restrict__ c0, const float* __restrict__ c1,
                     float* __restrict__ T1) {
  int id = blockIdx.x * 256 + threadIdx.x;           // 4096
  int d0 = id >> 9, d1 = (id >> 6) & 7, r2 = id & 63;
  float s = 0.f;
  for (int r1 = 0; r1 < 64; ++r1)
    s += c0[d0 * 64 + r1] * c1[(r1 * 8 + d1) * 64 + r2];
  T1[id] = s;
}

__global__ void k_t2(const float* __restrict__ T1, const float* __restrict__ c2,
                     float* __restrict__ T2) {
  int id = blockIdx.x * 256 + threadIdx.x;           // 32768
  int a = id >> 9, d2 = (id >> 6) & 7, r3 = id & 63;
  float s = 0.f;
  for (int r2 = 0; r2 < 64; ++r2)
    s += T1[a * 64 + r2] * c2[(r2 * 8 + d2) * 64 + r3];
  T2[id] = s;
}

__global__ void k_u(const float* __restrict__ T2, const float* __restrict__ c3,
                    float* __restrict__ U) {
  int id = blockIdx.x * 256 + threadIdx.x;           // 262144
  int p = id >> 9, d3 = (id >> 6) & 7, r4 = id & 63;
  float s = 0.f;
  for (int r3 = 0; r3 < 64; ++r3)
    s += T2[p * 64 + r3] * c3[(r3 * 8 + d3) * 64 + r4];
  U[id] = s;                                          // flat = (input index i)*64 + r4
}

__global__ void k_q2(const float* __restrict__ c6, const float* __restrict__ c7,
                     float* __restrict__ Q2) {
  int id = blockIdx.x * 256 + threadIdx.x;           // 4096
  int s5 = id >> 6, d6 = (id >> 3) & 7, d7 = id & 7;
  float s = 0.f;
  for (int s6 = 0; s6 < 64; ++s6)
    s += c6[(s5 * 8 + d6) * 64 + s6] * c7[s6 * 8 + d7];
  Q2[id] = s;
}

__global__ void k_q3(const float* __restrict__ c5, const float* __restrict__ Q2,
                     float* __restrict__ Q3) {
  int id = blockIdx.x * 256 + threadIdx.x;           // 32768
  int s4 = id >> 9, d5 = (id >> 6) & 7, j2 = id & 63;
  float s = 0.f;
  for (int s5 = 0; s5 < 64; ++s5)
    s += c5[(s4 * 8 + d5) * 64 + s5] * Q2[s5 * 64 + j2];
  Q3[id] = s;
}

__global__ void k_v(const float* __restrict__ c4, const float* __restrict__ Q3,
                    float* __restrict__ V) {
  int id = blockIdx.x * 256 + threadIdx.x;           // 262144
  int r4 = id >> 12, d4 = (id >> 9) & 7, jj = id & 511;
  float s = 0.f;
  for (int s4 = 0; s4 < 64; ++s4)
    s += c4[(r4 * 8 + d4) * 64 + s4] * Q3[s4 * 512 + jj];
  V[id] = s;                                          // flat = r4*4096 + (output index j)
}

// ---- pack U/V into per-lane WMMA B-operand layout (one coalesced b64 per lane) ----
// B operand for 16x16x4: lane l holds {B[k0, n], B[k0+1, n]},
//   n = nbase + (l&15), k0 = 4t + 2*(l>>4)
__global__ void k_packu(const float* __restrict__ U, float* __restrict__ Upk) {
  int id = blockIdx.x * 256 + threadIdx.x;           // 1000*4*32 = 128000
  int lane = id & 31, nb = (id >> 5) & 3, t = id >> 7;
  int col = nb * 16 + (lane & 15);
  int k0 = 4 * t + ((lane >> 4) << 1);
  Upk[2 * id]     = U[(size_t)k0 * 64 + col];
  Upk[2 * id + 1] = U[(size_t)(k0 + 1) * 64 + col];
}

__global__ void k_packv(const float* __restrict__ V, float* __restrict__ Vpk) {
  int id = blockIdx.x * 256 + threadIdx.x;           // 256*16*32 = 131072
  int lane = id & 31, t = (id >> 5) & 15, ntile = id >> 9;
  int col = ntile * 16 + (lane & 15);
  int k0 = 4 * t + ((lane >> 4) << 1);
  Vpk[2 * id]     = V[(size_t)k0 * 4096 + col];
  Vpk[2 * id + 1] = V[(size_t)(k0 + 1) * 4096 + col];
}

// ---------------- GEMM 1: Zp[c] += X[:, Kc] @ U[Kc, :]  (K split 4 ways) ----------------
// Block = 4 waves = the 4 K-chunks of one 16-row M-tile. 2048 waves total.
__global__ __launch_bounds__(128) void k_xu(const float* __restrict__ X,
                                            const float* __restrict__ Upk,
                                            float* __restrict__ Zp) {
  const int lane  = threadIdx.x & 31;
  const int wave  = threadIdx.x >> 5;                 // K-chunk 0..3
  const int mtile = blockIdx.x;                       // 0..511
  const int row   = mtile * 16 + (lane & 15);
  const int khalf = (lane >> 4) << 1;                 // 0 or 2
  const int t0    = wave * 250;                       // 250 K-steps per chunk
  const float* xptr = X + (size_t)row * 4000 + (size_t)t0 * 4 + khalf;
  const float* bptr = Upk + (size_t)t0 * 256 + (size_t)lane * 2;
  v8f a0{}, a1{}, a2{}, a3{};
  for (int t = 0; t < 250; ++t) {
    v2f av = __builtin_nontemporal_load((const v2f*)xptr);  // x streamed once
    xptr += 4;
    v2f b0 = *(const v2f*)(bptr);
    v2f b1 = *(const v2f*)(bptr + 64);
    v2f b2 = *(const v2f*)(bptr + 128);
    v2f b3 = *(const v2f*)(bptr + 192);
    bptr += 256;
    a0 = wmma4(av, b0, a0);
    a1 = wmma4(av, b1, a1);
    a2 = wmma4(av, b2, a2);
    a3 = wmma4(av, b3, a3);
  }
  // C layout: VGPR i -> row (i + 8*(lane>>4)), col lane&15
  const int h = lane >> 4, col = lane & 15;
  float* zp = Zp + (size_t)wave * 524288 +
              (size_t)(mtile * 16 + h * 8) * 64 + col;
#pragma unroll
  for (int i = 0; i < 8; ++i) {
    zp[(size_t)i * 64 + 0]  = a0[i];
    zp[(size_t)i * 64 + 16] = a1[i];
    zp[(size_t)i * 64 + 32] = a2[i];
    zp[(size_t)i * 64 + 48] = a3[i];
  }
}

// ---- reduce the 4 K-partials (fixed order => deterministic) into the packed
//      WMMA A-operand layout consumed by k_zv: lane l of (mtile,t) holds
//      {Z[row, k0], Z[row, k0+1]}, row = mtile*16+(l&15), k0 = 4t+2*(l>>4) ----
__global__ void k_reduce(const float* __restrict__ Zp, float* __restrict__ Zpk) {
  int id = blockIdx.x * 256 + threadIdx.x;           // 512*16*32 = 262144
  int lane = id & 31, t = (id >> 5) & 15, mtile = id >> 9;
  int row = mtile * 16 + (lane & 15);
  int k0 = 4 * t + ((lane >> 4) << 1);
  size_t base = (size_t)row * 64 + k0;
  float s0 = Zp[base]               + Zp[524288 + base];
  float s1 = Zp[base + 1]           + Zp[524288 + base + 1];
  s0 += Zp[2 * 524288 + base];      s1 += Zp[2 * 524288 + base + 1];
  s0 += Zp[3 * 524288 + base];      s1 += Zp[3 * 524288 + base + 1];
  Zpk[2 * id]     = s0;
  Zpk[2 * id + 1] = s1;
}

// ---------------- GEMM 2: OUT(8192x4096) = Z @ V + bias ----------------
// One wave: 16 rows x 64 cols; A (16x64 of Z) preloaded from packed layout
// (fully coalesced b64), K = 16 WMMA steps per 16x16 tile. 32768 waves.
__global__ __launch_bounds__(128) void k_zv(const float* __restrict__ Zpk,
                                            const float* __restrict__ Vpk,
                                            const float* __restrict__ bias,
                                            float* __restrict__ OUT) {
  const int lane = threadIdx.x & 31;
  const int wave = threadIdx.x >> 5;
  const int wid  = blockIdx.x * 4 + wave;             // 0..32767
  const int mtile = wid >> 6;                         // 0..511
  const int ngrp  = wid & 63;                         // 0..63
  const int h = lane >> 4, col = lane & 15;
  v2f a[16];
  const float* ap = Zpk + ((size_t)mtile * 16 * 32 + lane) * 2;
#pragma unroll
  for (int t = 0; t < 16; ++t) a[t] = *(const v2f*)(ap + t * 64);
#pragma unroll
  for (int nt = 0; nt < 4; ++nt) {
    const int ntile = ngrp * 4 + nt;
    const float* bp = Vpk + ((size_t)ntile * 16 * 32 + lane) * 2;
    v8f acc{};
#pragma unroll
    for (int t = 0; t < 16; ++t)
      acc = wmma4(a[t], *(const v2f*)(bp + t * 64), acc);
    const float bv = bias[ntile * 16 + col];
    float* op = OUT + (size_t)(mtile * 16 + h * 8) * 4096 + ntile * 16 + col;
#pragma unroll
    for (int i = 0; i < 8; ++i)
      __builtin_nontemporal_store(acc[i] + bv, op + (size_t)i * 4096);
  }
}

extern "C" void kernel_launch(void* const* d_in, const int* in_sizes, int n_in,
                              void* d_out, int out_size, void* d_ws, size_t ws_size,
                              hipStream_t stream) {
  const float* X    = (const float*)d_in[0];
  const float* c0   = (const float*)d_in[1];
  const float* c1   = (const float*)d_in[2];
  const float* c2   = (const float*)d_in[3];
  const float* c3   = (const float*)d_in[4];
  const float* c4   = (const float*)d_in[5];
  const float* c5   = (const float*)d_in[6];
  const float* c6   = (const float*)d_in[7];
  const float* c7   = (const float*)d_in[8];
  const float* bias = (const float*)d_in[9];

  float* ws  = (float*)d_ws;
  float* T1  = ws + WS_T1;
  float* T2  = ws + WS_T2;
  float* U   = ws + WS_U;
  float* Q2  = ws + WS_Q2;
  float* Q3  = ws + WS_Q3;
  float* V   = ws + WS_V;
  float* Upk = ws + WS_UPK;
  float* Vpk = ws + WS_VPK;
  float* Zp  = ws + WS_ZP;
  float* Zpk = ws + WS_ZPK;

  // compose TT cores into low-rank factors (deterministic, ~40 MFLOP total)
  k_t1<<<16, 256, 0, stream>>>(c0, c1, T1);
  k_t2<<<128, 256, 0, stream>>>(T1, c2, T2);
  k_u<<<1024, 256, 0, stream>>>(T2, c3, U);
  k_q2<<<16, 256, 0, stream>>>(c6, c7, Q2);
  k_q3<<<128, 256, 0, stream>>>(c5, Q2, Q3);
  k_v<<<1024, 256, 0, stream>>>(c4, Q3, V);
  k_packu<<<500, 256, 0, stream>>>(U, Upk);
  k_packv<<<512, 256, 0, stream>>>(V, Vpk);

  // Z partials = X @ U  (2048 waves: 512 M-tiles x 4 K-chunks)
  k_xu<<<512, 128, 0, stream>>>(X, Upk, Zp);
  // deterministic 4-way reduce into packed A-operand layout
  k_reduce<<<1024, 256, 0, stream>>>(Zp, Zpk);
  // OUT = Z @ V + bias  (32768 waves)
  k_zv<<<8192, 128, 0, stream>>>(Zpk, Vpk, bias, (float*)d_out);
}